// MiniMaxGateReference_10840497455874
// MI455X (gfx1250) — compile-verified
//
#include <hip/hip_runtime.h>

typedef __attribute__((ext_vector_type(2))) float v2f;
typedef __attribute__((ext_vector_type(4))) float v4f;
typedef __attribute__((ext_vector_type(8))) float v8f;

#define EXPERTS 64
#define TOPK    8
#define ROWS    32          // tokens per workgroup (two WMMA M-tiles per wave)
#define WAVES   8           // wave32 waves per workgroup (8-way K split)
#define THREADS (WAVES * 32)

// Two chained fp32 WMMAs covering 8 K-columns held in one B128 load per operand.
// Lane loads cols c..c+3 with c = kb + 4*(lane>>4): lo lanes cover kb..kb+3,
// hi lanes kb+4..kb+7. Slot mapping is identical for A and B, so accumulating
// .xy then .zw sums each of the 8 columns exactly once.
static __device__ inline v8f wmma_pair(v4f a, v4f b, v8f c) {
    v2f al = __builtin_shufflevector(a, a, 0, 1);
    v2f bl = __builtin_shufflevector(b, b, 0, 1);
    v2f ah = __builtin_shufflevector(a, a, 2, 3);
    v2f bh = __builtin_shufflevector(b, b, 2, 3);
    c = __builtin_amdgcn_wmma_f32_16x16x4_f32(false, al, false, bl, (short)0, c, false, false);
    c = __builtin_amdgcn_wmma_f32_16x16x4_f32(false, ah, false, bh, (short)0, c, false, false);
    return c;
}

__global__ __launch_bounds__(THREADS)
void moe_gate_fused_kernel(const float* __restrict__ x,
                           const float* __restrict__ gw,
                           const float* __restrict__ bias,
                           int*   __restrict__ out_idx,
                           float* __restrict__ out_w,
                           int D)
{
    __shared__ float s_part[WAVES][ROWS][EXPERTS];   // 64 KB K-split partials
    __shared__ float s_bias[EXPERTS];
    // Overlay after the reduction consumes s_part: score -> s_part[0], biased -> s_part[1]
    float (*s_score)[EXPERTS]  = s_part[0];
    float (*s_biased)[EXPERTS] = s_part[1];

    const int tid  = threadIdx.x;
    const int wave = tid >> 5;
    const int lane = tid & 31;
    const int l15  = lane & 15;
    const int lhi  = lane >> 4;

    if (tid < EXPERTS) s_bias[tid] = bias[tid];

    // ---- GEMM phase: 32x64 logits tile, fp32 WMMA 16x16x4, B128 operand loads ----
    const long long row0 = (long long)blockIdx.x * ROWS + l15;
    const float* xrow0 = x + row0 * (long long)D;            // M-tile 0: rows 0..15
    const float* xrow1 = xrow0 + 16 * (long long)D;          // M-tile 1: rows 16..31
    const float* w0 = gw + (long long)(l15 +  0) * D;
    const float* w1 = gw + (long long)(l15 + 16) * D;
    const float* w2 = gw + (long long)(l15 + 32) * D;
    const float* w3 = gw + (long long)(l15 + 48) * D;
    const int koff = 4 * lhi;

    v8f acc00 = {}, acc01 = {}, acc02 = {}, acc03 = {};
    v8f acc10 = {}, acc11 = {}, acc12 = {}, acc13 = {};

    const int kslice = D / WAVES;
    const int kbeg = wave * kslice;
    const int kend = kbeg + kslice;

    #pragma unroll 1
    for (int kb = kbeg; kb < kend; kb += 16) {
        const int c = kb + koff;
        // A operands: 2 M-tiles x 2 K-chunks (B128 each)
        v4f a0A = *(const v4f*)(xrow0 + c);
        v4f a0B = *(const v4f*)(xrow0 + c + 8);
        v4f a1A = *(const v4f*)(xrow1 + c);
        v4f a1B = *(const v4f*)(xrow1 + c + 8);
        // B operands: 4 N-tiles x 2 K-chunks (B128 each)
        v4f b0A = *(const v4f*)(w0 + c);
        v4f b0B = *(const v4f*)(w0 + c + 8);
        v4f b1A = *(const v4f*)(w1 + c);
        v4f b1B = *(const v4f*)(w1 + c + 8);
        v4f b2A = *(const v4f*)(w2 + c);
        v4f b2B = *(const v4f*)(w2 + c + 8);
        v4f b3A = *(const v4f*)(w3 + c);
        v4f b3B = *(const v4f*)(w3 + c + 8);

        acc00 = wmma_pair(a0A, b0A, acc00);  acc00 = wmma_pair(a0B, b0B, acc00);
        acc01 = wmma_pair(a0A, b1A, acc01);  acc01 = wmma_pair(a0B, b1B, acc01);
        acc02 = wmma_pair(a0A, b2A, acc02);  acc02 = wmma_pair(a0B, b2B, acc02);
        acc03 = wmma_pair(a0A, b3A, acc03);  acc03 = wmma_pair(a0B, b3B, acc03);
        acc10 = wmma_pair(a1A, b0A, acc10);  acc10 = wmma_pair(a1B, b0B, acc10);
        acc11 = wmma_pair(a1A, b1A, acc11);  acc11 = wmma_pair(a1B, b1B, acc11);
        acc12 = wmma_pair(a1A, b2A, acc12);  acc12 = wmma_pair(a1B, b2B, acc12);
        acc13 = wmma_pair(a1A, b3A, acc13);  acc13 = wmma_pair(a1B, b3B, acc13);
    }

    // C/D layout: VGPR r of lane l holds (M = r + 8*(l>>4), N = l&15)
    #pragma unroll
    for (int r = 0; r < 8; ++r) {
        const int m0 = r + 8 * lhi;
        const int m1 = m0 + 16;
        s_part[wave][m0][ 0 + l15] = acc00[r];
        s_part[wave][m0][16 + l15] = acc01[r];
        s_part[wave][m0][32 + l15] = acc02[r];
        s_part[wave][m0][48 + l15] = acc03[r];
        s_part[wave][m1][ 0 + l15] = acc10[r];
        s_part[wave][m1][16 + l15] = acc11[r];
        s_part[wave][m1][32 + l15] = acc12[r];
        s_part[wave][m1][48 + l15] = acc13[r];
    }
    __syncthreads();

    // ---- cross-wave K-reduction + sigmoid + bias (overlays results onto s_part) ----
    for (int i = tid; i < ROWS * EXPERTS; i += THREADS) {
        const int m = i >> 6;
        const int e = i & 63;
        float s = 0.f;
        #pragma unroll
        for (int w = 0; w < WAVES; ++w) s += s_part[w][m][e];
        const float sc = 1.0f / (1.0f + expf(-s));
        s_score[m][e]  = sc;              // s_part[0][m][e]
        s_biased[m][e] = sc + s_bias[e];  // s_part[1][m][e]
    }
    __syncthreads();

    // ---- top-8 per token (iterated argmax: descending, lowest index on ties) ----
    if (tid < ROWS) {
        const long long token = (long long)blockIdx.x * ROWS + tid;
        int   sel[TOPK];
        float sv[TOPK];
        float ssum = 0.f;
        #pragma unroll
        for (int k = 0; k < TOPK; ++k) {
            float best = -3.4e38f;
            int   bi = 0;
            for (int e = 0; e < EXPERTS; ++e) {
                const float v = s_biased[tid][e];
                if (v > best) { best = v; bi = e; }
            }
            s_biased[tid][bi] = -3.4e38f;
            sel[k] = bi;
            const float s = s_score[tid][bi];
            sv[k] = s;
            ssum += s;
        }
        const float inv = 1.0f / (ssum + 1e-20f);
        #pragma unroll
        for (int k = 0; k < TOPK; ++k) {
            out_idx[token * TOPK + k] = sel[k];
            out_w [token * TOPK + k] = sv[k] * inv;
        }
    }
}

extern "C" void kernel_launch(void* const* d_in, const int* in_sizes, int n_in,
                              void* d_out, int out_size, void* d_ws, size_t ws_size,
                              hipStream_t stream) {
    (void)n_in; (void)out_size; (void)d_ws; (void)ws_size;

    const float* x    = (const float*)d_in[0];   // [T, D] f32
    const float* gw   = (const float*)d_in[1];   // [E, D] f32
    const float* bias = (const float*)d_in[2];   // [E]    f32

    const int E = in_sizes[2];                   // 64
    const int D = in_sizes[1] / E;               // 4096
    const long long T = (long long)in_sizes[0] / D;  // 8192

    // Tuple output: indices [T,8] int32 first, then weights [T,8] f32.
    int*   out_idx = (int*)d_out;
    float* out_w   = (float*)((int*)d_out + T * TOPK);

    dim3 grid((unsigned)(T / ROWS));
    dim3 block(THREADS);
    moe_gate_fused_kernel<<<grid, block, 0, stream>>>(x, gw, bias, out_idx, out_w, D);
}